// DSTBranch_50319836840626
// MI455X (gfx1250) — compile-verified
//
#include <hip/hip_runtime.h>
#include <math.h>

typedef __attribute__((ext_vector_type(2))) float v2f;
typedef __attribute__((ext_vector_type(8))) float v8f;

#define NTHREADS 256
#define NWAVES   8
#define MCAP     112      // max gathered rows (M1 = 100 in practice), 7 x 16 tiles
#define ATTLD    112      // attention matrix leading dim
#define IDXCAP   128

// ---------------------------------------------------------------------------
// dilation-mask predicate, exactly matching the numpy float32 semantics:
// for a side^3 cube: keep cell iff d = sqrt(S) satisfies mod(d,dil)==0 or d==0,
// which (S integer <= 300) holds iff S is a perfect square whose root is a
// multiple of dil, or S == 0.  Non-cube N: arange(0, N, dil).
// ---------------------------------------------------------------------------
__device__ __forceinline__ bool dil_pred(int i, int dil, int side, int c) {
  if (side == 0) return (i % dil) == 0;
  int z  = i / (side * side);
  int y  = (i / side) % side;
  int xx = i % side;
  int S = (z - c) * (z - c) + (y - c) * (y - c) + (xx - c) * (xx - c);
  if (S == 0) return true;
  int r = 0;
  while (r * r < S) ++r;
  return (r * r == S) && (r % dil == 0);
}

// Parallel compaction of the index set into s_idx (ascending). Returns count
// as a wave-uniform (readfirstlane'd) scalar.
__device__ int make_idx(int N, int dil, int* s_idx, int* s_scan, int tid) {
  int side = 0;
  for (int s = 1; s <= 32; ++s)
    if (s * s * s == N) side = s;
  const int c = side >> 1;
  const int chunk = (N + NTHREADS - 1) / NTHREADS;
  const int lo = tid * chunk;
  const int hi = (lo + chunk < N) ? (lo + chunk) : N;

  int cnt = 0;
  for (int i = lo; i < hi; ++i)
    if (dil_pred(i, dil, side, c)) ++cnt;
  s_scan[tid] = cnt;
  __syncthreads();
  if (tid == 0) {
    int acc = 0;
    for (int j = 0; j < NTHREADS; ++j) { int v = s_scan[j]; s_scan[j] = acc; acc += v; }
    s_scan[NTHREADS] = acc;
  }
  __syncthreads();
  int off = s_scan[tid];
  for (int i = lo; i < hi; ++i)
    if (dil_pred(i, dil, side, c) && off < IDXCAP) s_idx[off++] = i;
  __syncthreads();
  int total = s_scan[NTHREADS];            // identical across all threads
  total = __builtin_amdgcn_readfirstlane(total);  // force SGPR-uniform
  return total > MCAP ? MCAP : total;
}

// ---------------------------------------------------------------------------
// Tiled f32 WMMA GEMM:  C[Mt*16 x Nt*16] = A[Mt*16 x Ksteps*4] * B
// A in LDS (row-major, stride lda).  B source selected by BMODE:
//   0: global row-major   B[k][n] = Bg[k*ldb + n]
//   1: global transposed  B[k][n] = Bg[n*ldb + k]
//   2: LDS row-major      B[k][n] = Bl[k*ldb + n]
//   3: LDS transposed     B[k][n] = Bl[n*ldb + k]
// VGPR layouts per CDNA5 ISA 7.12.2 (wave32, 16x16x4 f32).
// The tile loop is made structurally wave-uniform (scalar bounds + scalar
// induction variable) so WMMA always executes with EXEC = all-1s.
// ---------------------------------------------------------------------------
template <int BMODE>
__device__ __forceinline__ void wmma_gemm(const float* A, int lda,
                                          const float* Bg, const float* Bl,
                                          int ldb, float* C, int ldc,
                                          int Mt, int Nt, int Ksteps, int tid) {
  const int wave = __builtin_amdgcn_readfirstlane(tid >> 5);  // uniform wave id
  const int lane = tid & 31;
  const int half = lane >> 4;   // 0: K={0,1}, 1: K={2,3}
  const int lr   = lane & 15;
  const int nt   = Mt * Nt;
  for (int t = wave; t < nt; t += NWAVES) {
    const int i = t / Nt;
    const int j = t % Nt;
    v8f acc = {0.f, 0.f, 0.f, 0.f, 0.f, 0.f, 0.f, 0.f};
    const float* arow = A + (i * 16 + lr) * lda;
    const int col = j * 16 + lr;
    for (int ks = 0; ks < Ksteps; ++ks) {
      const int k0 = ks * 4 + 2 * half;
      v2f a, bfrag;
      a.x = arow[k0];
      a.y = arow[k0 + 1];
      if (BMODE == 0) {
        bfrag.x = Bg[k0 * ldb + col];       bfrag.y = Bg[(k0 + 1) * ldb + col];
      } else if (BMODE == 1) {
        bfrag.x = Bg[col * ldb + k0];       bfrag.y = Bg[col * ldb + k0 + 1];
      } else if (BMODE == 2) {
        bfrag.x = Bl[k0 * ldb + col];       bfrag.y = Bl[(k0 + 1) * ldb + col];
      } else {
        bfrag.x = Bl[col * ldb + k0];       bfrag.y = Bl[col * ldb + k0 + 1];
      }
      acc = __builtin_amdgcn_wmma_f32_16x16x4_f32(false, a, false, bfrag,
                                                  (short)0, acc, false, false);
    }
    float* cc = C + col;
#pragma unroll
    for (int r = 0; r < 8; ++r)
      cc[(i * 16 + r + 8 * half) * ldc] = acc[r];
  }
}

// ---------------------------------------------------------------------------
// Fully fused DST branch: conv(gathered) -> 3x {QKV, QK^T, softmax, AV, proj}
// One workgroup (8 wave32) per batch element; all intermediates in LDS.
// The residual in each DST block uses the *post-attention* tensor, so only
// the dilation-gathered rows of each block input are ever needed: the 8000
// token sequence collapses to 100 -> 25 -> 5 rows, and the conv is evaluated
// only at the 100 gathered grid positions.
// ---------------------------------------------------------------------------
extern "C" __global__ void __launch_bounds__(NTHREADS)
dst_branch_fused(const float* __restrict__ x, const float* __restrict__ conv_w,
                 const float* __restrict__ conv_b, const float* __restrict__ pos,
                 const float* __restrict__ qkv_w, const float* __restrict__ qkv_b,
                 const float* __restrict__ proj_w, const float* __restrict__ proj_b,
                 float* __restrict__ out) {
  // LDS layout (floats): ATT[0..12544) (IN aliases its first 7168),
  // Q[12544), K[19712), V[26880), O[34048) .. 41216  => 164,864 B
  __shared__ float smem[41216];
  __shared__ int s_idx[IDXCAP];
  __shared__ int s_scan[NTHREADS + 1];
  float* s_ATT = smem;
  float* s_IN  = smem;            // alias: IN is dead before ATT is written
  float* s_Q   = smem + 12544;
  float* s_K   = smem + 19712;
  float* s_V   = smem + 26880;
  float* s_O   = smem + 34048;

  const int tid = (int)threadIdx.x;
  const int b   = (int)blockIdx.x;

  // ---- index set for block 1 over the 20^3 conv grid, dilation 2 ----
  int M    = make_idx(8000, 2, s_idx, s_scan, tid);
  int Mt   = (M + 15) >> 4;
  int Mt16 = Mt << 4;

  // ---- stage gathered 4x4x4 conv patches into s_Q (zero-padded rows) ----
  for (int t = tid; t < Mt16 * 64; t += NTHREADS) {
    const int n = t >> 6, e = t & 63;
    float v = 0.0f;
    if (n < M) {
      const int fi = s_idx[n];
      const int z = fi / 400, y = (fi / 20) % 20, xx = fi % 20;
      const int dz = e >> 4, dy = (e >> 2) & 3, dx = e & 3;
      v = x[(size_t)b * 512000 + (size_t)(4 * z + dz) * 6400 +
            (4 * y + dy) * 80 + (4 * xx + dx)];
    }
    s_Q[t] = v;
  }
  __syncthreads();

  // conv as GEMM: IN = patches @ conv_w^T   (B[k][n] = conv_w[n*64+k])
  wmma_gemm<1>(s_Q, 64, conv_w, nullptr, 64, s_IN, 64, Mt, 4, 16, tid);
  __syncthreads();

  // + conv bias + pos_embed (valid rows only; pad rows stay exactly 0)
  for (int t = tid; t < M * 64; t += NTHREADS) {
    const int n = t >> 6, cch = t & 63;
    s_IN[t] += conv_b[cch] + pos[cch * 8000 + s_idx[n]];
  }
  __syncthreads();

  const int dils[3] = {2, 4, 6};
  for (int blk = 0; blk < 3; ++blk) {
    if (blk > 0) {
      // gather idx rows of previous block output into IN, zero-pad tiles
      const int Mn = make_idx(M, dils[blk], s_idx, s_scan, tid);
      const int Mnt16 = ((Mn + 15) >> 4) << 4;
      for (int t = tid; t < Mnt16 * 64; t += NTHREADS) {
        const int r = t >> 6, cch = t & 63;
        s_IN[t] = (r < Mn) ? s_O[s_idx[r] * 64 + cch] : 0.0f;
      }
      M = Mn; Mt = (M + 15) >> 4; Mt16 = Mt << 4;
      __syncthreads();
    }
    const float* Wq = qkv_w + blk * 64 * 192;
    const float* Bq = qkv_b + blk * 192;
    const float* Wp = proj_w + blk * 64 * 64;
    const float* Bp = proj_b + blk * 64;

    // QKV GEMMs (column slices of the [64,192] weight)
    wmma_gemm<0>(s_IN, 64, Wq + 0,   nullptr, 192, s_Q, 64, Mt, 4, 16, tid);
    wmma_gemm<0>(s_IN, 64, Wq + 64,  nullptr, 192, s_K, 64, Mt, 4, 16, tid);
    wmma_gemm<0>(s_IN, 64, Wq + 128, nullptr, 192, s_V, 64, Mt, 4, 16, tid);
    __syncthreads();
    for (int t = tid; t < M * 64; t += NTHREADS) {
      const int cch = t & 63;
      s_Q[t] += Bq[cch];
      s_K[t] += Bq[64 + cch];
      s_V[t] += Bq[128 + cch];
    }
    __syncthreads();

    // ATT = Q @ K^T  (clobbers the dead IN alias)
    wmma_gemm<3>(s_Q, 64, nullptr, s_K, 64, s_ATT, ATTLD, Mt, Mt, 16, tid);
    __syncthreads();

    // row softmax with 1/sqrt(64) scaling; pad cols stay 0 (K pad rows = 0)
    for (int r = tid; r < M; r += NTHREADS) {
      float* row = s_ATT + r * ATTLD;
      float mx = -3.4e38f;
      for (int j2 = 0; j2 < M; ++j2) {
        const float lv = row[j2] * 0.125f;
        mx = lv > mx ? lv : mx;
      }
      float sum = 0.f;
      for (int j2 = 0; j2 < M; ++j2) {
        const float e = expf(row[j2] * 0.125f - mx);
        row[j2] = e;
        sum += e;
      }
      const float inv = 1.0f / sum;
      for (int j2 = 0; j2 < M; ++j2) row[j2] *= inv;
    }
    __syncthreads();

    // O = ATT @ V   (K-dim = Mt16; pad cols of ATT are exact zeros)
    wmma_gemm<2>(s_ATT, ATTLD, nullptr, s_V, 64, s_O, 64, Mt, 4, Mt * 4, tid);
    __syncthreads();

    // P = O @ Wp  -> s_Q (reuse), then O += P + proj bias (residual on O)
    wmma_gemm<0>(s_O, 64, Wp, nullptr, 64, s_Q, 64, Mt, 4, 16, tid);
    __syncthreads();
    for (int t = tid; t < M * 64; t += NTHREADS)
      s_O[t] += s_Q[t] + Bp[t & 63];
    __syncthreads();
  }

  // final output: [B, M3, 64]
  for (int t = tid; t < M * 64; t += NTHREADS)
    out[(size_t)b * (M * 64) + t] = s_O[t];
}

extern "C" void kernel_launch(void* const* d_in, const int* in_sizes, int n_in,
                              void* d_out, int out_size, void* d_ws, size_t ws_size,
                              hipStream_t stream) {
  const float* x      = (const float*)d_in[0];
  const float* conv_w = (const float*)d_in[1];
  const float* conv_b = (const float*)d_in[2];
  const float* pos    = (const float*)d_in[3];
  const float* qkv_w  = (const float*)d_in[4];
  const float* qkv_b  = (const float*)d_in[5];
  const float* proj_w = (const float*)d_in[6];
  const float* proj_b = (const float*)d_in[7];
  const int B = in_sizes[0] / (80 * 80 * 80);
  dst_branch_fused<<<dim3(B), dim3(NTHREADS), 0, stream>>>(
      x, conv_w, conv_b, pos, qkv_w, qkv_b, proj_w, proj_b, (float*)d_out);
}